// CategorySpecificLinear_60473139528224
// MI455X (gfx1250) — compile-verified
//
#include <hip/hip_runtime.h>

#if __has_builtin(__builtin_amdgcn_global_load_async_to_lds_b128)
#define USE_ASYNC_LDS 1
#else
#define USE_ASYNC_LDS 0
#endif

typedef __attribute__((ext_vector_type(2))) float v2f;
typedef __attribute__((ext_vector_type(4))) float f4;
typedef __attribute__((ext_vector_type(8))) float v8f;
typedef __attribute__((ext_vector_type(4))) int   v4i;

#define NSAMP 4096
#define MROWS 16
#define KDIM 256
#define HDIM 256

#define XS 260            // LDS row stride (floats) for x: 260%64=4 -> conflict-free A b64 reads
#define WS 272            // LDS row stride (floats) for W: 2*272%64=32 -> half-waves hit disjoint banks
#define CHUNK 16          // K rows of W staged per step
#define NCHUNK (KDIM / CHUNK)

__device__ __forceinline__ void wait_async0() {
#if __has_builtin(__builtin_amdgcn_s_wait_asynccnt)
    __builtin_amdgcn_s_wait_asynccnt(0);
#else
    asm volatile("s_wait_asynccnt 0x0" ::: "memory");
#endif
}

#if USE_ASYNC_LDS
typedef __attribute__((address_space(1))) v4i gas_v4i;   // global
typedef __attribute__((address_space(3))) v4i las_v4i;   // LDS

// Direct global -> LDS 128-bit async copy (no VGPR round trip), ASYNCcnt-tracked.
__device__ __forceinline__ void async_copy_b128(const float* g, float* l) {
    __builtin_amdgcn_global_load_async_to_lds_b128(
        (gas_v4i*)g,
        (las_v4i*)l,
        0, 0);
}
#endif

__global__ __launch_bounds__(256) void cat_linear_wmma_kernel(
    const float* __restrict__ x, const int* __restrict__ cat_ids,
    const float* __restrict__ W, const float* __restrict__ b,
    float* __restrict__ out)
{
    __shared__ float lds_x[MROWS * XS];          // 16.6 KB
    __shared__ float lds_w[2][CHUNK * WS];       // 2 x 17.4 KB

    const int n   = blockIdx.x;
    const int tid = threadIdx.x;
    const int cat = cat_ids[n];

    const float* xg = x + (size_t)n * MROWS * KDIM;
    const float* wg = W + (size_t)cat * KDIM * HDIM;

#if USE_ASYNC_LDS
    // ---- async stage: x[n] (16x256) and W chunk 0, straight into LDS ----
    #pragma unroll
    for (int j = 0; j < 4; ++j) {
        int f  = tid + 256 * j;      // float4 linear index (64 float4 per row)
        int r  = f >> 6;
        int c4 = f & 63;
        async_copy_b128(xg + r * KDIM + c4 * 4, &lds_x[r * XS + c4 * 4]);
    }
    #pragma unroll
    for (int j = 0; j < 4; ++j) {
        int f  = tid + 256 * j;
        int r  = f >> 6;
        int c4 = f & 63;
        async_copy_b128(wg + r * KDIM + c4 * 4, &lds_w[0][r * WS + c4 * 4]);
    }
    wait_async0();
    __syncthreads();
#else
    // ---- fallback: synchronous staging through VGPRs ----
    #pragma unroll
    for (int j = 0; j < 4; ++j) {
        int f  = tid + 256 * j;
        int r  = f >> 6;
        int c4 = f & 63;
        f4 v = *(const f4*)(xg + r * KDIM + c4 * 4);
        *(f4*)(&lds_x[r * XS + c4 * 4]) = v;
    }
    f4 wreg[4];
    #pragma unroll
    for (int j = 0; j < 4; ++j) {
        int f  = tid + 256 * j;
        int r  = f >> 6;
        int c4 = f & 63;
        wreg[j] = *(const f4*)(wg + r * KDIM + c4 * 4);
    }
    #pragma unroll
    for (int j = 0; j < 4; ++j) {
        int f  = tid + 256 * j;
        int r  = f >> 6;
        int c4 = f & 63;
        *(f4*)(&lds_w[0][r * WS + c4 * 4]) = wreg[j];
    }
    __syncthreads();
#endif

    const int wave = tid >> 5;
    const int lane = tid & 31;
    const int ln   = lane & 15;      // matrix row (A) / column (B,C,D) within tile
    const int hi   = lane >> 4;      // half-wave select
    const int t0   = wave * 2;       // first of the 2 column tiles owned by this wave

    v8f acc0 = {};
    v8f acc1 = {};

    for (int c = 0; c < NCHUNK; ++c) {
        const int cur = c & 1;

#if USE_ASYNC_LDS
        // Issue async fill of the other buffer. Safe: buf[1-cur] was last READ
        // in iteration c-1, which completed before the barrier ending c-1.
        if (c + 1 < NCHUNK) {
            const float* wn = wg + (size_t)(c + 1) * CHUNK * KDIM;
            #pragma unroll
            for (int j = 0; j < 4; ++j) {
                int f  = tid + 256 * j;
                int r  = f >> 6;
                int c4 = f & 63;
                async_copy_b128(wn + r * KDIM + c4 * 4,
                                &lds_w[1 - cur][r * WS + c4 * 4]);
            }
        }
#else
        if (c + 1 < NCHUNK) {
            const float* wn = wg + (size_t)(c + 1) * CHUNK * KDIM;
            #pragma unroll
            for (int j = 0; j < 4; ++j) {
                int f  = tid + 256 * j;
                int r  = f >> 6;
                int c4 = f & 63;
                wreg[j] = *(const f4*)(wn + r * KDIM + c4 * 4);
            }
        }
#endif

        // 4 WMMA k-steps of K=4 over this 16-row chunk, for both tiles
        #pragma unroll
        for (int kk = 0; kk < 4; ++kk) {
            const int kbase = c * CHUNK + kk * 4;
            // A fragment: lane ln holds row M=ln; VGPR0/1 = K = kbase+2*hi, +1
            v2f a = *(const v2f*)(&lds_x[ln * XS + kbase + 2 * hi]);

            // B fragments: VGPR0 = row kb, VGPR1 = row kb+1, column = tile*16+ln
            const int kb = kk * 4 + 2 * hi;
            v2f b0, b1;
            b0.x = lds_w[cur][(kb    ) * WS + t0 * 16 + ln];
            b0.y = lds_w[cur][(kb + 1) * WS + t0 * 16 + ln];
            b1.x = lds_w[cur][(kb    ) * WS + (t0 + 1) * 16 + ln];
            b1.y = lds_w[cur][(kb + 1) * WS + (t0 + 1) * 16 + ln];

            acc0 = __builtin_amdgcn_wmma_f32_16x16x4_f32(
                false, a, false, b0, (short)0, acc0, false, false);
            acc1 = __builtin_amdgcn_wmma_f32_16x16x4_f32(
                false, a, false, b1, (short)0, acc1, false, false);
        }

#if USE_ASYNC_LDS
        wait_async0();       // this wave's async fills of buf[1-cur] landed
        __syncthreads();     // all waves: reads of buf[cur] done, fills visible
#else
        __syncthreads();     // all waves done reading lds_w[cur]
        if (c + 1 < NCHUNK) {
            #pragma unroll
            for (int j = 0; j < 4; ++j) {
                int f  = tid + 256 * j;
                int r  = f >> 6;
                int c4 = f & 63;
                *(f4*)(&lds_w[1 - cur][r * WS + c4 * 4]) = wreg[j];
            }
            __syncthreads();
        }
#endif
    }

    // ---- epilogue: bias add + store. C/D layout: VGPR v -> row v+8*hi, col tile*16+ln ----
    const float* bg = b + (size_t)cat * HDIM;
    float* og = out + (size_t)n * MROWS * HDIM;
    const float bias0 = bg[t0 * 16 + ln];
    const float bias1 = bg[(t0 + 1) * 16 + ln];

    #pragma unroll
    for (int v = 0; v < 8; ++v) {
        int row = v + 8 * hi;
        og[row * HDIM + t0 * 16 + ln]       = acc0[v] + bias0;
        og[row * HDIM + (t0 + 1) * 16 + ln] = acc1[v] + bias1;
    }
}

extern "C" void kernel_launch(void* const* d_in, const int* in_sizes, int n_in,
                              void* d_out, int out_size, void* d_ws, size_t ws_size,
                              hipStream_t stream) {
    const float* x       = (const float*)d_in[0];
    const int*   cat_ids = (const int*)  d_in[1];
    const float* W       = (const float*)d_in[2];
    const float* b       = (const float*)d_in[3];
    float*       out     = (float*)d_out;

    dim3 grid(NSAMP);
    dim3 block(256);
    hipLaunchKernelGGL(cat_linear_wmma_kernel, grid, block, 0, stream,
                       x, cat_ids, W, b, out);
}